// GraphAttentionLayer_34531537059969
// MI455X (gfx1250) — compile-verified
//
#include <hip/hip_runtime.h>
#include <math.h>

#define ALPHA   0.2f
#define N_NODES 4096
#define IN_FEAT 512
#define NHEAD   4
#define FDIM    64
#define HF      256   // NHEAD * FDIM

typedef __attribute__((ext_vector_type(2))) float v2f;
typedef __attribute__((ext_vector_type(8))) float v8f;

__device__ __forceinline__ float lrelu(float x) {
    // alpha < 1  =>  leaky_relu(x) == max(x, alpha*x)
    return fmaxf(x, ALPHA * x);
}

// ---------------------------------------------------------------------------
// Kernel 1: h = x @ W   (4096x512 @ 512x256), f32 WMMA 16x16x4
// block = 128 threads (4 waves); wave w computes a 16x16 tile at col n0+w*16
// ---------------------------------------------------------------------------
__global__ __launch_bounds__(128) void gat_gemm1(const float* __restrict__ x,
                                                 const float* __restrict__ W,
                                                 float* __restrict__ h) {
    const int lane = threadIdx.x & 31;
    const int wave = threadIdx.x >> 5;
    const int i0   = blockIdx.x * 16;              // M tile
    const int n0   = blockIdx.y * 64 + wave * 16;  // N tile for this wave
    const int m    = lane & 15;                    // A row / B-C col within tile
    const int koff = (lane >> 4) << 1;             // 0 or 2 (half-wave K split)

    const float* xrow = x + (size_t)(i0 + m) * IN_FEAT;
    v8f acc = {};
    for (int k0 = 0; k0 < IN_FEAT; k0 += 4) {
        const int ka = k0 + koff;
        v2f a2, b2;
        a2.x = xrow[ka];                           // A: K=koff   (VGPR0)
        a2.y = xrow[ka + 1];                       // A: K=koff+1 (VGPR1)
        b2.x = W[(size_t)ka * HF + n0 + m];        // B: K=koff,   N=m
        b2.y = W[(size_t)(ka + 1) * HF + n0 + m];  // B: K=koff+1, N=m
        acc = __builtin_amdgcn_wmma_f32_16x16x4_f32(false, a2, false, b2,
                                                    (short)0, acc, false, false);
    }
    const int rbase = (lane >> 4) << 3;            // C/D: M = v + 8*(lane/16)
#pragma unroll
    for (int v = 0; v < 8; ++v)
        h[(size_t)(i0 + rbase + v) * HF + n0 + m] = acc[v];
}

// ---------------------------------------------------------------------------
// Kernel 2: s[n,hd] = h[n,hd,:].a_src[hd]   t[n,hd] = h[n,hd,:].a_dst[hd]
// one wave32 per (node, head); 2 elements per lane + xor-shuffle reduce
// ---------------------------------------------------------------------------
__global__ __launch_bounds__(128) void gat_st(const float* __restrict__ h,
                                              const float* __restrict__ a,
                                              float* __restrict__ s,
                                              float* __restrict__ t) {
    const int lane = threadIdx.x & 31;
    const int gw   = blockIdx.x * 4 + (threadIdx.x >> 5);  // 0..16383
    const int n    = gw >> 2;
    const int hd   = gw & 3;

    const float* hrow = h + (size_t)n * HF + hd * FDIM;
    const float* asrc = a + hd * 2 * FDIM;     // a[hd, 0:64, 0]
    const float* adst = asrc + FDIM;           // a[hd, 64:128, 0]

    float sv = hrow[lane] * asrc[lane] + hrow[lane + 32] * asrc[lane + 32];
    float tv = hrow[lane] * adst[lane] + hrow[lane + 32] * adst[lane + 32];
#pragma unroll
    for (int off = 16; off > 0; off >>= 1) {
        sv += __shfl_xor(sv, off, 32);
        tv += __shfl_xor(tv, off, 32);
    }
    if (lane == 0) {
        s[n * NHEAD + hd] = sv;
        t[n * NHEAD + hd] = tv;
    }
}

// ---------------------------------------------------------------------------
// Kernel 3: tmax[hd] = max_j t[j,hd]  (leaky_relu monotone => exact softmax max)
// ---------------------------------------------------------------------------
__global__ __launch_bounds__(256) void gat_tmax(const float* __restrict__ t,
                                                float* __restrict__ tmax) {
    __shared__ float red[256];
    const int hd = blockIdx.x;
    float m = -INFINITY;
    for (int j = threadIdx.x; j < N_NODES; j += 256)
        m = fmaxf(m, t[j * NHEAD + hd]);
    red[threadIdx.x] = m;
    __syncthreads();
    for (int off = 128; off > 0; off >>= 1) {
        if (threadIdx.x < off)
            red[threadIdx.x] = fmaxf(red[threadIdx.x], red[threadIdx.x + off]);
        __syncthreads();
    }
    if (threadIdx.x == 0) tmax[hd] = red[0];
}

// ---------------------------------------------------------------------------
// Kernel 4: out[i, hd*64+f] = sum_j softmax_j(lrelu(s_i+t_j)) * h[j, hd*64+f]
// block = (16-row tile, head); 4 waves split the j (K) range; each wave holds
// the full 16x64 accumulator (4 WMMA tiles); P built in registers, Z alongside;
// cross-wave reduce through LDS, then scale by 1/Z.
// ---------------------------------------------------------------------------
__global__ __launch_bounds__(128) void gat_attn(const float* __restrict__ h,
                                                const float* __restrict__ s,
                                                const float* __restrict__ t,
                                                const float* __restrict__ tmax,
                                                float* __restrict__ out) {
    __shared__ float cred[4][16][64];   // 16 KB partial C tiles
    __shared__ float zred[4][32];       // per-lane Z partials
    __shared__ float zrow[16];          // final per-row denominators

    const int lane = threadIdx.x & 31;
    const int wave = threadIdx.x >> 5;
    const int i0   = blockIdx.x * 16;
    const int hd   = blockIdx.y;
    const int m    = lane & 15;
    const int koff = (lane >> 4) << 1;

    const float sm   = s[(i0 + m) * NHEAD + hd];
    const float mrow = lrelu(sm + tmax[hd]);       // exact row max of e
    const float* hh  = h + hd * FDIM;              // h[j, hd*64 + c] = hh[j*HF + c]

    v8f acc0 = {}, acc1 = {}, acc2 = {}, acc3 = {};
    float zpart = 0.f;

    const int jbeg = wave * (N_NODES / 4);
    const int jend = jbeg + (N_NODES / 4);
    for (int j0 = jbeg; j0 < jend; j0 += 4) {
        const int ja = j0 + koff;
        const float pa = __expf(lrelu(sm + t[ja * NHEAD + hd]) - mrow);
        const float pb = __expf(lrelu(sm + t[(ja + 1) * NHEAD + hd]) - mrow);
        zpart += pa + pb;
        v2f a2; a2.x = pa; a2.y = pb;              // P tile: M=m, K=koff..koff+1

        const float* hja = hh + (size_t)ja * HF;   // B row K=koff
        const float* hjb = hja + HF;               // B row K=koff+1
        v2f b0, b1, b2, b3;
        b0.x = hja[m];      b0.y = hjb[m];
        b1.x = hja[16 + m]; b1.y = hjb[16 + m];
        b2.x = hja[32 + m]; b2.y = hjb[32 + m];
        b3.x = hja[48 + m]; b3.y = hjb[48 + m];

        acc0 = __builtin_amdgcn_wmma_f32_16x16x4_f32(false, a2, false, b0, (short)0, acc0, false, false);
        acc1 = __builtin_amdgcn_wmma_f32_16x16x4_f32(false, a2, false, b1, (short)0, acc1, false, false);
        acc2 = __builtin_amdgcn_wmma_f32_16x16x4_f32(false, a2, false, b2, (short)0, acc2, false, false);
        acc3 = __builtin_amdgcn_wmma_f32_16x16x4_f32(false, a2, false, b3, (short)0, acc3, false, false);
    }

    const int rbase = (lane >> 4) << 3;            // C/D row = v + 8*(lane/16)
#pragma unroll
    for (int v = 0; v < 8; ++v) {
        cred[wave][rbase + v][m]      = acc0[v];
        cred[wave][rbase + v][16 + m] = acc1[v];
        cred[wave][rbase + v][32 + m] = acc2[v];
        cred[wave][rbase + v][48 + m] = acc3[v];
    }
    zred[wave][lane] = zpart;
    __syncthreads();

    if (threadIdx.x < 16) {
        float z = 0.f;
#pragma unroll
        for (int w = 0; w < 4; ++w)
            z += zred[w][threadIdx.x] + zred[w][threadIdx.x + 16];
        zrow[threadIdx.x] = z;
    }
    __syncthreads();

    for (int e = threadIdx.x; e < 16 * 64; e += 128) {
        const int rr = e >> 6, cc = e & 63;
        const float v = cred[0][rr][cc] + cred[1][rr][cc] +
                        cred[2][rr][cc] + cred[3][rr][cc];
        out[(size_t)(i0 + rr) * HF + hd * FDIM + cc] = v / zrow[rr];
    }
}

// ---------------------------------------------------------------------------
extern "C" void kernel_launch(void* const* d_in, const int* in_sizes, int n_in,
                              void* d_out, int out_size, void* d_ws, size_t ws_size,
                              hipStream_t stream) {
    const float* x = (const float*)d_in[0];
    // d_in[1] = adj (int32): present in setup_inputs but UNUSED by reference()
    const float* W = (const float*)d_in[2];
    const float* a = (const float*)d_in[3];
    float* out = (float*)d_out;

    char*  ws   = (char*)d_ws;
    float* h    = (float*)ws;                                    // 4096*256 f32 (4 MB)
    float* s    = (float*)(ws + (size_t)N_NODES * HF * sizeof(float));
    float* t    = s + N_NODES * NHEAD;
    float* tmax = t + N_NODES * NHEAD;

    dim3 g1(N_NODES / 16, HF / 64);                 // (256, 4)
    gat_gemm1<<<g1, 128, 0, stream>>>(x, W, h);

    gat_st<<<N_NODES, 128, 0, stream>>>(h, a, s, t); // 4096 blocks * 4 waves = 16384 (n,hd)

    gat_tmax<<<NHEAD, 256, 0, stream>>>(t, tmax);

    dim3 g4(N_NODES / 16, NHEAD);                   // (256, 4)
    gat_attn<<<g4, 128, 0, stream>>>(h, s, t, tmax, out);
}